// myGCN_5978594476291
// MI455X (gfx1250) — compile-verified
//
#include <hip/hip_runtime.h>

#define NNODES 20000
#define NEDGES 320000
#define DIM    256
#define NGRAPH 128

typedef __attribute__((ext_vector_type(2))) float v2f;
typedef __attribute__((ext_vector_type(8))) float v8f;

// ---------------- degree / normalization ----------------
__global__ void k_deg_init(float* __restrict__ deg) {
    int i = blockIdx.x * blockDim.x + threadIdx.x;
    if (i < NNODES) deg[i] = 1.0f;                 // self-loop contributes 1
}

__global__ void k_deg_count(const int* __restrict__ dst, float* __restrict__ deg) {
    int e = blockIdx.x * blockDim.x + threadIdx.x;
    if (e < NEDGES) atomicAdd(&deg[dst[e]], 1.0f);
}

__global__ void k_dis(const float* __restrict__ deg, float* __restrict__ dis) {
    int i = blockIdx.x * blockDim.x + threadIdx.x;
    if (i < NNODES) dis[i] = rsqrtf(deg[i]);       // deg >= 1 always (self-loop)
}

// ---------------- fp32 WMMA 16x16 tile: acc = A[m0:16, :256] * B[:256, n0:16]
// Wave32; A 16x4 fragment: lane = M%16, K = 2*(lane/16)+j (contiguous -> b64).
// B 4x16 fragment: lane = N%16, K = 2*(lane/16)+j. C/D: VGPR r -> M = r+8*(lane/16).
__device__ __forceinline__ v8f gemm_tile(const float* __restrict__ A,
                                         const float* __restrict__ B,
                                         int m0, int n0, int half, int lq) {
    const float* arow = A + (size_t)(m0 + lq) * DIM + half * 2;
    const float* bcol = B + (size_t)(half * 2) * DIM + n0 + lq;
    v8f acc = {0.f, 0.f, 0.f, 0.f, 0.f, 0.f, 0.f, 0.f};
#pragma unroll 8
    for (int k = 0; k < DIM; k += 4) {
        v2f a = *(const v2f*)(arow + k);
        v2f b;
        b.x = bcol[k * DIM];
        b.y = bcol[k * DIM + DIM];
        acc = __builtin_amdgcn_wmma_f32_16x16x4_f32(false, a, false, b,
                                                    (short)0, acc, false, false);
    }
    return acc;
}

// conv GEMM: writes h = A*B and agg = h * dis^2 (self-loop term fused)
__global__ void __launch_bounds__(256)
k_gemm_conv(const float* __restrict__ A, const float* __restrict__ B,
            const float* __restrict__ dis,
            float* __restrict__ h, float* __restrict__ agg) {
    const int lane = threadIdx.x & 31;
    const int wave = threadIdx.x >> 5;
    const int half = lane >> 4, lq = lane & 15;
    const int m0 = blockIdx.x * 32 + (wave >> 2) * 16;
    const int n0 = blockIdx.y * 64 + (wave & 3) * 16;
    v8f acc = gemm_tile(A, B, m0, n0, half, lq);
#pragma unroll
    for (int r = 0; r < 8; ++r) {
        int m = m0 + r + half * 8;
        size_t idx = (size_t)m * DIM + n0 + lq;
        float v = acc[r];
        float d = dis[m];
        h[idx]   = v;
        agg[idx] = v * d * d;
    }
}

// residual GEMM with fused pooling: pooled[batch[m]] += acc + agg2 + b2 + blin
__global__ void __launch_bounds__(256)
k_gemm_pool(const float* __restrict__ A, const float* __restrict__ B,
            const float* __restrict__ agg2, const float* __restrict__ b2,
            const float* __restrict__ blin, const int* __restrict__ batch,
            float* __restrict__ pooled) {
    const int lane = threadIdx.x & 31;
    const int wave = threadIdx.x >> 5;
    const int half = lane >> 4, lq = lane & 15;
    const int m0 = blockIdx.x * 32 + (wave >> 2) * 16;
    const int n0 = blockIdx.y * 64 + (wave & 3) * 16;
    v8f acc = gemm_tile(A, B, m0, n0, half, lq);
    const int f = n0 + lq;
    const float bb = b2[f] + blin[f];
#pragma unroll
    for (int r = 0; r < 8; ++r) {
        int m = m0 + r + half * 8;
        float s = acc[r] + agg2[(size_t)m * DIM + f] + bb;
        atomicAdd(&pooled[batch[m] * DIM + f], s);
    }
}

// ---------------- edge scatter: agg[dst] += h[src] * dis[src]*dis[dst] ------
// One wave per edge; edge id forced uniform (SGPR) so index/degree loads are
// scalar; each lane covers 8 features via two coalesced float4 gathers.
__global__ void __launch_bounds__(256)
k_edge_scatter(const int* __restrict__ src, const int* __restrict__ dst,
               const float* __restrict__ dis, const float* __restrict__ h,
               float* __restrict__ agg) {
    int t = blockIdx.x * 256 + threadIdx.x;
    int e = __builtin_amdgcn_readfirstlane(t >> 5);   // wave-uniform edge id
    int lane = threadIdx.x & 31;
    if (e < NEDGES) {
        int s = src[e], d = dst[e];                   // scalar (SMEM) loads
        float nrm = dis[s] * dis[d];
        const float* hp = h + (size_t)s * DIM;
        float* op = agg + (size_t)d * DIM;
        int f0 = lane << 2;
        float4 a = *(const float4*)(hp + f0);
        float4 b = *(const float4*)(hp + f0 + 128);
        atomicAdd(op + f0 + 0, a.x * nrm);
        atomicAdd(op + f0 + 1, a.y * nrm);
        atomicAdd(op + f0 + 2, a.z * nrm);
        atomicAdd(op + f0 + 3, a.w * nrm);
        atomicAdd(op + f0 + 128 + 0, b.x * nrm);
        atomicAdd(op + f0 + 128 + 1, b.y * nrm);
        atomicAdd(op + f0 + 128 + 2, b.z * nrm);
        atomicAdd(op + f0 + 128 + 3, b.w * nrm);
    }
}

// ---------------- bias + relu (after conv1 aggregation) ----------------
__global__ void k_bias_relu(const float* __restrict__ agg, const float* __restrict__ b,
                            float* __restrict__ out) {
    int idx = blockIdx.x * blockDim.x + threadIdx.x;
    if (idx < NNODES * DIM) {
        float v = agg[idx] + b[idx & (DIM - 1)];
        out[idx] = v > 0.f ? v : 0.f;
    }
}

// ---------------- pooling helpers ----------------
__global__ void k_zero_pool(float* __restrict__ pooled, float* __restrict__ counts) {
    int idx = blockIdx.x * blockDim.x + threadIdx.x;
    if (idx < NGRAPH * DIM) pooled[idx] = 0.f;
    if (idx < NGRAPH) counts[idx] = 0.f;
}

__global__ void k_counts(const int* __restrict__ batch, float* __restrict__ counts) {
    int i = blockIdx.x * blockDim.x + threadIdx.x;
    if (i < NNODES) atomicAdd(&counts[batch[i]], 1.0f);
}

__global__ void k_out(const float* __restrict__ pooled, const float* __restrict__ counts,
                      float* __restrict__ out) {
    int idx = blockIdx.x * blockDim.x + threadIdx.x;
    if (idx < NGRAPH * DIM) {
        float c = counts[idx >> 8];
        out[idx] = pooled[idx] / fmaxf(c, 1.0f);
    }
}

// ---------------- launch ----------------
extern "C" void kernel_launch(void* const* d_in, const int* in_sizes, int n_in,
                              void* d_out, int out_size, void* d_ws, size_t ws_size,
                              hipStream_t stream) {
    const float* x     = (const float*)d_in[0];
    const float* W1    = (const float*)d_in[1];
    const float* b1    = (const float*)d_in[2];
    const float* W2    = (const float*)d_in[3];
    const float* b2    = (const float*)d_in[4];
    const float* Wlin  = (const float*)d_in[5];
    const float* blin  = (const float*)d_in[6];
    const int*   eidx  = (const int*)d_in[7];   // [2, E] int32
    const int*   batch = (const int*)d_in[8];   // [N]   int32
    const int* src = eidx;
    const int* dst = eidx + NEDGES;

    float* deg    = (float*)d_ws;
    float* dis    = deg + NNODES;
    float* h      = dis + NNODES;                       // N*D
    float* agg    = h   + (size_t)NNODES * DIM;         // N*D
    float* xb     = agg + (size_t)NNODES * DIM;         // N*D (x1)
    float* pooled = xb  + (size_t)NNODES * DIM;         // G*D
    float* counts = pooled + NGRAPH * DIM;              // G
    float* out = (float*)d_out;

    dim3 b256(256);
    dim3 ggrid(NNODES / 32, DIM / 64);                  // 625 x 4, exact
    const int gND  = (NNODES * DIM + 255) / 256;
    const int gE32 = (NEDGES * 32) / 256;               // 40000, exact

    // normalization
    k_deg_init <<<(NNODES + 255) / 256, b256, 0, stream>>>(deg);
    k_deg_count<<<(NEDGES + 255) / 256, b256, 0, stream>>>(dst, deg);
    k_dis      <<<(NNODES + 255) / 256, b256, 0, stream>>>(deg, dis);

    // conv1: h = x@W1 (+fused self-loop agg); scatter; x1 = relu(agg + b1)
    k_gemm_conv   <<<ggrid, b256, 0, stream>>>(x, W1, dis, h, agg);
    k_edge_scatter<<<gE32, b256, 0, stream>>>(src, dst, dis, h, agg);
    k_bias_relu   <<<gND, b256, 0, stream>>>(agg, b1, xb);

    // conv2: h = x1@W2 (+fused self-loop agg); scatter (b2 folded into pooling)
    k_gemm_conv   <<<ggrid, b256, 0, stream>>>(xb, W2, dis, h, agg);
    k_edge_scatter<<<gE32, b256, 0, stream>>>(src, dst, dis, h, agg);

    // pooling: zero, count, then residual GEMM with fused pooled atomics
    k_zero_pool<<<(NGRAPH * DIM + 255) / 256, b256, 0, stream>>>(pooled, counts);
    k_counts   <<<(NNODES + 255) / 256, b256, 0, stream>>>(batch, counts);
    k_gemm_pool<<<ggrid, b256, 0, stream>>>(x, Wlin, agg, b2, blin, batch, pooled);
    k_out      <<<(NGRAPH * DIM + 255) / 256, b256, 0, stream>>>(pooled, counts, out);
}